// NeuroPlasticComponent_27805618274543
// MI455X (gfx1250) — compile-verified
//
#include <hip/hip_runtime.h>
#include <hip/hip_bf16.h>

typedef __attribute__((ext_vector_type(2))) float v2f;
typedef __attribute__((ext_vector_type(8))) float v8f;

#define D_MODEL 768
#define D_FFN   3072
#define RANK    16
#define SEQ     64
#define MOD_SCALE 0.1f
#define FT_PER_WAVE 4          // f-tiles (16 cols each) per wave

// ---------------------------------------------------------------------------
// Kernel 1: modulation[s][r] = MOD_SCALE * tanh( sum_d attn[s][d] * A[d][r] )
// 64*16 = 1024 independent dot products of length 768. Negligible cost.
// ---------------------------------------------------------------------------
__global__ __launch_bounds__(256)
void np_modulation_kernel(const float* __restrict__ attn,
                          const float* __restrict__ A,
                          float* __restrict__ mod) {
    int t = blockIdx.x * blockDim.x + threadIdx.x;   // 0..1023
    int s = t >> 4;           // sequence position
    int r = t & 15;           // rank index
    const float* arow = attn + s * D_MODEL;
    float acc = 0.0f;
#pragma unroll 8
    for (int d = 0; d < D_MODEL; ++d)
        acc = fmaf(arow[d], A[d * RANK + r], acc);
    mod[t] = MOD_SCALE * tanhf(acc);
}

// ---------------------------------------------------------------------------
// Kernel 2: delta_w[s] (768x3072) = (A * diag(mod[s])) @ B  via fp32 WMMA.
// One wave computes a 16x64 output strip (4 adjacent 16x16 tiles):
//   - A' operand (A * diag(mod[s])) loaded/scaled ONCE, reused for 4 WMMAs
//     per K-group -> 16 accumulating V_WMMA_F32_16X16X4_F32 total
//   - 32 b32 stores, all sharing one address VGPR pair via immediate
//     offsets (i*12288 + j*64 bytes, within signed-24-bit IOFFSET)
// Store-bandwidth bound; this layout minimizes non-store issue slots.
//
// Operand VGPR layouts (ISA 7.12.2, 32-bit 16x4 A / 4x16 B):
//   A: lane = 16*h + m  holds {A'[m][2h+4g], A'[m][2h+4g+1]}  (v2f)
//   B: lane = 16*h + n  holds {B[2h+4g][n],  B[2h+4g+1][n]}   (v2f)
//   D: VGPR i, lane 16*h+n  -> element (d0 + i + 8h, f0 + n)
// ---------------------------------------------------------------------------
__global__ __launch_bounds__(256)
void np_deltaw_kernel(const float* __restrict__ A,
                      const float* __restrict__ B,
                      const float* __restrict__ mod,
                      float* __restrict__ out) {
    const int NFG = D_FFN / (16 * FT_PER_WAVE);   // 48 f-strips
    const int NDT = D_MODEL / 16;                 // 48 d-tiles

    const int lane = threadIdx.x & 31;
    const int wave = threadIdx.x >> 5;

    int task = blockIdx.x * (blockDim.x >> 5) + wave;   // < 64*48*48
    int fg  = task % NFG;
    int tmp = task / NFG;
    int dt  = tmp % NDT;
    int s   = tmp / NDT;

    const int d0 = dt * 16;
    const int f0 = fg * (16 * FT_PER_WAVE);
    const int lm = lane & 15;        // m (for A) / n (for B/D)
    const int h  = lane >> 4;        // lane-half selects K sub-pair
    const int c  = h * 2;            // K offset within each group of 4

    const float* Arow = A   + (d0 + lm) * RANK + c;      // A[d0+lm][c + ...]
    const float* Mrow = mod + s * RANK + c;              // mod[s][c + ...]
    const float* Bcol = B   + c * D_FFN + f0 + lm;       // B[c + ...][f0+lm]

    v8f acc0 = {}, acc1 = {}, acc2 = {}, acc3 = {};
#pragma unroll
    for (int g = 0; g < 4; ++g) {
        // A' = A * diag(mod[s]) : fold modulation into the A operand; this
        // operand is shared by the 4 WMMAs of this K-group.
        v2f a = *(const v2f*)(Arow + 4 * g);
        v2f m = *(const v2f*)(Mrow + 4 * g);
        a.x *= m.x;
        a.y *= m.y;

        v2f b0, b1, b2, b3;
        b0.x = Bcol[(4 * g + 0) * D_FFN +  0];
        b0.y = Bcol[(4 * g + 1) * D_FFN +  0];
        b1.x = Bcol[(4 * g + 0) * D_FFN + 16];
        b1.y = Bcol[(4 * g + 1) * D_FFN + 16];
        b2.x = Bcol[(4 * g + 0) * D_FFN + 32];
        b2.y = Bcol[(4 * g + 1) * D_FFN + 32];
        b3.x = Bcol[(4 * g + 0) * D_FFN + 48];
        b3.y = Bcol[(4 * g + 1) * D_FFN + 48];

        // (neg_a, A, neg_b, B, c_mod, C, reuse_a, reuse_b)
        acc0 = __builtin_amdgcn_wmma_f32_16x16x4_f32(false, a, false, b0,
                                                     (short)0, acc0, false, false);
        acc1 = __builtin_amdgcn_wmma_f32_16x16x4_f32(false, a, false, b1,
                                                     (short)0, acc1, false, false);
        acc2 = __builtin_amdgcn_wmma_f32_16x16x4_f32(false, a, false, b2,
                                                     (short)0, acc2, false, false);
        acc3 = __builtin_amdgcn_wmma_f32_16x16x4_f32(false, a, false, b3,
                                                     (short)0, acc3, false, false);
    }

    // Store 16x64 strip: VGPR i covers rows d0+i (lanes 0-15) and d0+i+8
    // (lanes 16-31); j selects the 16-column sub-tile. One address pair,
    // 32 stores with immediate offsets.
    float* obase = out + ((size_t)s * D_MODEL + (size_t)(d0 + h * 8)) * D_FFN
                       + (size_t)(f0 + lm);
#pragma unroll
    for (int i = 0; i < 8; ++i) {
        obase[(size_t)i * D_FFN +  0] = acc0[i];
        obase[(size_t)i * D_FFN + 16] = acc1[i];
        obase[(size_t)i * D_FFN + 32] = acc2[i];
        obase[(size_t)i * D_FFN + 48] = acc3[i];
    }
}

// ---------------------------------------------------------------------------
extern "C" void kernel_launch(void* const* d_in, const int* in_sizes, int n_in,
                              void* d_out, int out_size, void* d_ws, size_t ws_size,
                              hipStream_t stream) {
    const float* attn = (const float*)d_in[0];   // (1, 64, 768)
    const float* A    = (const float*)d_in[1];   // (768, 16)
    const float* B    = (const float*)d_in[2];   // (16, 3072)
    float*       out  = (float*)d_out;           // (1, 64, 768, 3072)
    float*       mod  = (float*)d_ws;            // 64*16 floats scratch

    // 1) modulation: 1024 threads total
    np_modulation_kernel<<<4, 256, 0, stream>>>(attn, A, mod);

    // 2) delta_w: one wave per 16x64 strip, 8 waves per block
    const int tasks = SEQ * (D_MODEL / 16) * (D_FFN / (16 * FT_PER_WAVE)); // 147456
    const int waves_per_block = 256 / 32;                                  // 8
    np_deltaw_kernel<<<tasks / waves_per_block, 256, 0, stream>>>(A, B, mod, out);
}